// DenseAttentionMultiHead_26345329393765
// MI455X (gfx1250) — compile-verified
//
#include <hip/hip_runtime.h>

// f32 WMMA operand types (CDNA5 16x16x4 f32: A/B = 2 VGPRs, C/D = 8 VGPRs)
typedef __attribute__((ext_vector_type(2))) float v2f;
typedef __attribute__((ext_vector_type(8))) float v8f;

#define WMMA_F32(a, b, c) \
    __builtin_amdgcn_wmma_f32_16x16x4_f32(false, (a), false, (b), (short)0, (c), false, false)

// Problem constants (fixed by the reference)
//   X:   [B=4][S=2048][H=1024] f32
//   W:   [H=1024][H=1024] f32  (F.linear weight: [out,in])
//   out: [B][S][H] f32
// NH=16 heads, HD=64.
// Algebra (no softmax => reassociate):
//   out[:,hblk] = X @ (W_h^T @ G_{b,h}),   G_{b,h} = X_h^T X_h   (64x64)
// FLOPs ~22 GF instead of ~155 GF; no [B,h,S,S] intermediate.

// ---------------------------------------------------------------------------
// Kernel 1: partial G.  Gpart[ks][b][h] = sum_{k in slice ks} x_k^T x_k
// grid: x = bh (64), y = ksplit (4). Block = 128 threads (4 waves).
// The 16x64 X_h K-tile is staged once in LDS (row stride 80 => lanes 0-15 and
// 16-31 hit disjoint bank halves); A and B fragments are both LDS row reads.
// ---------------------------------------------------------------------------
__global__ void __launch_bounds__(128)
g_kernel(const float* __restrict__ X, float* __restrict__ Gpart) {
    __shared__ float Xt[16][80];

    const int bh   = blockIdx.x;           // 0..63
    const int ks   = blockIdx.y;           // 0..3  (K slice of 512)
    const int b    = bh >> 4;
    const int h    = bh & 15;
    const int wave = threadIdx.x >> 5;
    const int lane = threadIdx.x & 31;
    const int l15  = lane & 15;
    const int khalf = (lane >> 4) * 2;     // 0 for lanes 0-15, 2 for lanes 16-31
    const float* __restrict__ Xh = X + (size_t)b * 2048u * 1024u + h * 64;

    const int m0 = wave * 16;
    v8f acc[4];
#pragma unroll
    for (int t = 0; t < 4; ++t) acc[t] = (v8f)0.0f;

    // staging map: 128 threads x 8 floats = 16x64 tile
    const int srow = threadIdx.x >> 3;        // 0..15
    const int scol = (threadIdx.x & 7) * 8;   // 0,8,...,56

    const int kend = ks * 512 + 512;
    for (int k0 = ks * 512; k0 < kend; k0 += 16) {
        const float* src = Xh + (size_t)(k0 + srow) * 1024u + scol;
        float4 v0 = *(const float4*)(src);
        float4 v1 = *(const float4*)(src + 4);
        *(float2*)&Xt[srow][scol + 0] = make_float2(v0.x, v0.y);
        *(float2*)&Xt[srow][scol + 2] = make_float2(v0.z, v0.w);
        *(float2*)&Xt[srow][scol + 4] = make_float2(v1.x, v1.y);
        *(float2*)&Xt[srow][scol + 6] = make_float2(v1.z, v1.w);
        __syncthreads();

#pragma unroll
        for (int kk = 0; kk < 16; kk += 4) {
            const int rA = kk + khalf;
            v2f a;
            a.x = Xt[rA][m0 + l15];
            a.y = Xt[rA + 1][m0 + l15];
#pragma unroll
            for (int t = 0; t < 4; ++t) {
                v2f bb;
                bb.x = Xt[rA][t * 16 + l15];
                bb.y = Xt[rA + 1][t * 16 + l15];
                acc[t] = WMMA_F32(a, bb, acc[t]);
            }
        }
        __syncthreads();
    }

    // C/D layout: m = r + 8*(lane>=16), n = lane&15
    float* __restrict__ Gp = Gpart + ((size_t)ks * 64u + bh) * 64u * 64u;
    const int rowAdd = (lane >> 4) * 8;
#pragma unroll
    for (int t = 0; t < 4; ++t)
#pragma unroll
        for (int r = 0; r < 8; ++r)
            Gp[(m0 + r + rowAdd) * 64 + t * 16 + l15] = acc[t][r];
}

// ---------------------------------------------------------------------------
// Kernel 1b: deterministic split-K reduction  G = sum_ks Gpart[ks]
// ---------------------------------------------------------------------------
__global__ void __launch_bounds__(256)
g_reduce(const float* __restrict__ Gpart, float* __restrict__ G) {
    const int i = blockIdx.x * 256 + threadIdx.x;    // 0..262143
    const size_t SL = (size_t)64 * 64 * 64;          // one slice
    G[i] = ((Gpart[i] + Gpart[SL + i]) + Gpart[2 * SL + i]) + Gpart[3 * SL + i];
}

// ---------------------------------------------------------------------------
// Kernel 2: W'[b][k][h*64+n] = sum_d W[h*64+d][k] * G[b][h][d][n]
// M=1024 (k), N=64, K=64 per (b,h). A = W^T slice (lane-contiguous), B = G.
// grid: x = m-block (16, 64 rows each), y = bh (64). 4 waves/block.
// ---------------------------------------------------------------------------
__global__ void __launch_bounds__(128)
wprime_kernel(const float* __restrict__ W, const float* __restrict__ G,
              float* __restrict__ Wp) {
    const int bh   = blockIdx.y;
    const int b    = bh >> 4;
    const int h    = bh & 15;
    const int wave = threadIdx.x >> 5;
    const int lane = threadIdx.x & 31;
    const int l15  = lane & 15;
    const int khalf = (lane >> 4) * 2;
    const int m0 = blockIdx.x * 64 + wave * 16;

    const float* __restrict__ Wh = W + (size_t)(h * 64) * 1024u; // W[h*64+d][c]
    const float* __restrict__ Gp = G + (size_t)bh * 64u * 64u;

    v8f acc[4];
#pragma unroll
    for (int t = 0; t < 4; ++t) acc[t] = (v8f)0.0f;

#pragma unroll 4
    for (int d0 = 0; d0 < 64; d0 += 4) {
        const int rA = d0 + khalf;
        v2f a;
        a.x = Wh[(size_t)rA * 1024u + m0 + l15];
        a.y = Wh[(size_t)(rA + 1) * 1024u + m0 + l15];
#pragma unroll
        for (int t = 0; t < 4; ++t) {
            v2f bb;
            bb.x = Gp[rA * 64 + t * 16 + l15];
            bb.y = Gp[(rA + 1) * 64 + t * 16 + l15];
            acc[t] = WMMA_F32(a, bb, acc[t]);
        }
    }

    float* __restrict__ Wpb = Wp + (size_t)b * 1024u * 1024u;
    const int rowAdd = (lane >> 4) * 8;
    const int ncol0 = h * 64;
#pragma unroll
    for (int t = 0; t < 4; ++t)
#pragma unroll
        for (int r = 0; r < 8; ++r)
            Wpb[(size_t)(m0 + r + rowAdd) * 1024u + ncol0 + t * 16 + l15] = acc[t][r];
}

// ---------------------------------------------------------------------------
// Kernel 3: out[b] = X[b] @ W'[b]   (2048 x 1024 x 1024 per batch)
// grid: x = s-tile (16, 128 rows), y = n-block (16, 64 cols), z = b (4).
// 256 threads = 8 waves; wave w owns rows [16w,16w+16) x all 64 cols.
// A (X) staged in LDS (stride 18: 8B-aligned, distinct banks for col reads).
// B (W') staged in LDS once per block (was 8x-redundant global reads),
// stride 80 => lanes 0-15 / 16-31 read disjoint bank halves.
// ---------------------------------------------------------------------------
__global__ void __launch_bounds__(256)
out_kernel(const float* __restrict__ X, const float* __restrict__ Wp,
           float* __restrict__ out) {
    __shared__ float Xs[128][18];
    __shared__ float Ws[16][80];

    const int b      = blockIdx.z;
    const int s_base = blockIdx.x * 128;
    const int n_base = blockIdx.y * 64;
    const int wave   = threadIdx.x >> 5;
    const int lane   = threadIdx.x & 31;
    const int l15    = lane & 15;
    const int khalf  = (lane >> 4) * 2;
    const int m0     = wave * 16;

    const float* __restrict__ Xb  = X + (size_t)b * 2048u * 1024u;
    const float* __restrict__ Wpb = Wp + (size_t)b * 1024u * 1024u;

    v8f acc[4];
#pragma unroll
    for (int t = 0; t < 4; ++t) acc[t] = (v8f)0.0f;

    // X staging: 256 threads x 8 floats = 128x16 tile
    const int ldr = threadIdx.x >> 1;          // 0..127
    const int ldc = (threadIdx.x & 1) * 8;     // 0 or 8
    // W' staging: 256 threads x 4 floats = 16x64 tile
    const int wrow = threadIdx.x >> 4;         // 0..15
    const int wcol = (threadIdx.x & 15) * 4;   // 0..60

    for (int k0 = 0; k0 < 1024; k0 += 16) {
        {   // Xs[r][c] = X[b][s_base+r][k0+c]
            const float* src = &Xb[(size_t)(s_base + ldr) * 1024u + k0 + ldc];
            float4 v0 = *(const float4*)(src);
            float4 v1 = *(const float4*)(src + 4);
            *(float2*)&Xs[ldr][ldc + 0] = make_float2(v0.x, v0.y);
            *(float2*)&Xs[ldr][ldc + 2] = make_float2(v0.z, v0.w);
            *(float2*)&Xs[ldr][ldc + 4] = make_float2(v1.x, v1.y);
            *(float2*)&Xs[ldr][ldc + 6] = make_float2(v1.z, v1.w);
        }
        {   // Ws[r][c] = W'[b][k0+r][n_base+c]
            const float* wsrc = &Wpb[(size_t)(k0 + wrow) * 1024u + n_base + wcol];
            float4 wv = *(const float4*)(wsrc);
            *(float2*)&Ws[wrow][wcol + 0] = make_float2(wv.x, wv.y);
            *(float2*)&Ws[wrow][wcol + 2] = make_float2(wv.z, wv.w);
        }
        __syncthreads();

#pragma unroll
        for (int kk = 0; kk < 16; kk += 4) {
            const int rK = kk + khalf;
            v2f a;
            a.x = Xs[m0 + l15][rK];
            a.y = Xs[m0 + l15][rK + 1];
#pragma unroll
            for (int t = 0; t < 4; ++t) {
                v2f bb;
                bb.x = Ws[rK][t * 16 + l15];
                bb.y = Ws[rK + 1][t * 16 + l15];
                acc[t] = WMMA_F32(a, bb, acc[t]);
            }
        }
        __syncthreads();
    }

    float* __restrict__ ob = out + (size_t)b * 2048u * 1024u;
    const int rowAdd = (lane >> 4) * 8;
#pragma unroll
    for (int t = 0; t < 4; ++t)
#pragma unroll
        for (int r = 0; r < 8; ++r)
            ob[(size_t)(s_base + m0 + r + rowAdd) * 1024u + n_base + t * 16 + l15] =
                acc[t][r];
}

// ---------------------------------------------------------------------------
extern "C" void kernel_launch(void* const* d_in, const int* in_sizes, int n_in,
                              void* d_out, int out_size, void* d_ws, size_t ws_size,
                              hipStream_t stream) {
    (void)in_sizes; (void)n_in; (void)out_size; (void)ws_size;
    const float* X = (const float*)d_in[0];   // [4,2048,1024]
    const float* W = (const float*)d_in[1];   // [1024,1024]
    float* out = (float*)d_out;               // [4,2048,1024]

    // ws layout (floats): Gpart 4*64*64*64 (4MB) | G 64*64*64 (1MB) | W' 4*1024*1024 (16MB)
    float* Gpart = (float*)d_ws;
    float* G     = Gpart + (size_t)4 * 64 * 64 * 64;
    float* Wp    = G + (size_t)64 * 64 * 64;

    g_kernel<<<dim3(64, 4), dim3(128), 0, stream>>>(X, Gpart);
    g_reduce<<<dim3(1024), dim3(256), 0, stream>>>(Gpart, G);
    wprime_kernel<<<dim3(16, 64), dim3(128), 0, stream>>>(W, G, Wp);
    out_kernel<<<dim3(16, 16, 4), dim3(256), 0, stream>>>(X, Wp, out);
}